// DUQClassifier_72919954752033
// MI455X (gfx1250) — compile-verified
//
#include <hip/hip_runtime.h>

// ---- problem constants (match reference) ----
#define BATCH 4096
#define FEAT  1024      // K (j)
#define CDIM  512       // centroid dim (m)
#define NCLS  100       // classes (n)

// ---- tiling ----
#define BT   64         // batch rows per workgroup
#define MCH  128        // m rows per chunk (8 x 16-wide tiles)
#define KB   64         // K block staged in LDS
#define LSTRIDE 68      // padded LDS row stride in floats (kills 64-bank conflicts, keeps 16B align)

typedef __attribute__((ext_vector_type(2))) float v2f;
typedef __attribute__((ext_vector_type(8))) float v8f;
typedef __attribute__((ext_vector_type(4))) int   v4i;

// ---- CDNA5 async global->LDS path (guarded: falls back to VGPR staging if absent) ----
#if __has_builtin(__builtin_amdgcn_global_load_async_to_lds_b128) && \
    __has_builtin(__builtin_amdgcn_s_wait_asynccnt)
#define USE_ASYNC_LDS 1
#else
#define USE_ASYNC_LDS 0
#endif

__device__ __forceinline__ void stage16(const float* __restrict__ g, float* l) {
#if USE_ASYNC_LDS
    // GLOBAL_LOAD_ASYNC_TO_LDS_B128: 16B per lane, ASYNCcnt-tracked, no VGPR data.
    // Signature (from probe): (int4 global* src, int4 lds* dst, imm, imm)
    __builtin_amdgcn_global_load_async_to_lds_b128(
        (__attribute__((address_space(1))) v4i*)g,
        (__attribute__((address_space(3))) v4i*)l,
        0, 0);
#else
    *(float4*)l = *(const float4*)g;
#endif
}

__device__ __forceinline__ void stage_commit() {
#if USE_ASYNC_LDS
    __builtin_amdgcn_s_wait_asynccnt(0);   // own wave's async LDS writes landed
#endif
    __syncthreads();                        // everyone's writes visible
}

__global__ __launch_bounds__(256)
void duq_rbf_kernel(const float* __restrict__ x,      // [BATCH][FEAT]
                    const float* __restrict__ w,      // [CDIM][NCLS][FEAT]
                    const float* __restrict__ csum,   // [CDIM][NCLS]
                    const float* __restrict__ cnum,   // [NCLS]
                    float* __restrict__ out)          // conf [BATCH][NCLS] then logc [BATCH][NCLS]
{
    __shared__ float xs[BT  * LSTRIDE];   // ~17 KB
    __shared__ float ws[MCH * LSTRIDE];   // ~34 KB
    __shared__ float acc[BT];

    const int n    = blockIdx.y;            // class
    const int b0   = blockIdx.x * BT;       // batch base
    const int tid  = threadIdx.x;
    const int lane = tid & 31;              // wave32
    const int wave = tid >> 5;              // 0..7
    // 2x2 output tiles per wave: waves arranged 2 (row groups) x 4 (col groups)
    const int wr   = wave & 1;              // tile-row pair: rows {2*wr, 2*wr+1}   (4 x 16 = 64 batch)
    const int wc   = wave >> 1;             // tile-col pair: cols {2*wc, 2*wc+1}   (8 x 16 = 128 m)
    const int hl   = lane & 15;             // lane within 16-half
    const int hi   = lane >> 4;             // half select

    if (tid < BT) acc[tid] = 0.0f;

    const float inv_num = 1.0f / cnum[n];

    // staging assignments
    const int xr = tid >> 2;                // x row 0..63
    const int xc = (tid & 3) * 16;          // x col base (4 x float4 each)
    const int wrw = tid >> 1;               // w row 0..127
    const int wcb = (tid & 1) * 32;         // w col base (8 x float4 each)

    for (int mc = 0; mc < CDIM / MCH; ++mc) {
        v8f cacc[2][2] = {};

        for (int kb = 0; kb < FEAT / KB; ++kb) {
            __syncthreads();   // previous iteration's fragment reads complete

            // stage x tile: 64 rows x 64 floats (4 async B128 per thread)
            {
                const float* src = x + (size_t)(b0 + xr) * FEAT + kb * KB + xc;
                float*       dst = xs + xr * LSTRIDE + xc;
#pragma unroll
                for (int q = 0; q < 4; ++q)
                    stage16(src + 4 * q, dst + 4 * q);
            }
            // stage w tile: 128 rows x 64 floats (8 async B128 per thread)
            {
                const size_t wrow = (size_t)((mc * MCH + wrw) * NCLS + n) * FEAT;
                const float* src = w + wrow + kb * KB + wcb;
                float*       dst = ws + wrw * LSTRIDE + wcb;
#pragma unroll
                for (int q = 0; q < 8; ++q)
                    stage16(src + 4 * q, dst + 4 * q);
                if (kb + 1 < FEAT / KB)
                    __builtin_prefetch(w + wrow + (kb + 1) * KB + wcb, 0, 0);  // global_prefetch_b8
            }
            stage_commit();

            // K loop: 16 WMMA k-steps of 4; 2x2 tiles reuse both A and B fragments
#pragma unroll
            for (int k4 = 0; k4 < KB / 4; ++k4) {
                const int kbase = k4 * 4 + hi * 2;   // lanes 0-15: K,K+1 ; lanes 16-31: K+2,K+3

                v2f a[2], b[2];
#pragma unroll
                for (int jr = 0; jr < 2; ++jr) {
                    const float* ap = xs + ((wr * 2 + jr) * 16 + hl) * LSTRIDE + kbase;
                    a[jr].x = ap[0]; a[jr].y = ap[1];
                }
#pragma unroll
                for (int jc = 0; jc < 2; ++jc) {
                    const float* bp = ws + ((wc * 2 + jc) * 16 + hl) * LSTRIDE + kbase;
                    b[jc].x = bp[0]; b[jc].y = bp[1];
                }
#pragma unroll
                for (int jr = 0; jr < 2; ++jr)
#pragma unroll
                    for (int jc = 0; jc < 2; ++jc)
                        cacc[jr][jc] = __builtin_amdgcn_wmma_f32_16x16x4_f32(
                            false, a[jr], false, b[jc], (short)0, cacc[jr][jc], false, false);
            }
        }

        // fused RBF epilogue:
        // C layout: VGPR i -> lanes 0-15: (M=i, N=hl); lanes 16-31: (M=i+8, N=hl)
#pragma unroll
        for (int jr = 0; jr < 2; ++jr) {
#pragma unroll
            for (int jc = 0; jc < 2; ++jc) {
                const int m = mc * MCH + (wc * 2 + jc) * 16 + hl;
                const float cen = csum[m * NCLS + n] * inv_num;
#pragma unroll
                for (int i = 0; i < 8; ++i) {
                    const int brow = (wr * 2 + jr) * 16 + i + hi * 8;
                    const float d = cacc[jr][jc][i] - cen;
                    float s = d * d;
                    s += __shfl_xor(s, 1, 32);
                    s += __shfl_xor(s, 2, 32);
                    s += __shfl_xor(s, 4, 32);
                    s += __shfl_xor(s, 8, 32);        // sum over the 16 m's in this half
                    if (hl == 0) atomicAdd(&acc[brow], s);   // ds_add_f32
                }
            }
        }
    }

    __syncthreads();
    if (tid < BT) {
        const float sq   = acc[tid] * (1.0f / (float)CDIM);   // mean over centroid dim
        const float logc = -sq * 50.0f;                       // 1/(2*0.1^2) = 50
        const float conf = __expf(logc);
        const int b = b0 + tid;
        out[(size_t)b * NCLS + n] = conf;                                  // confidences
        out[(size_t)BATCH * NCLS + (size_t)b * NCLS + n] = logc;           // log_confidences
    }
}

extern "C" void kernel_launch(void* const* d_in, const int* in_sizes, int n_in,
                              void* d_out, int out_size, void* d_ws, size_t ws_size,
                              hipStream_t stream) {
    const float* x    = (const float*)d_in[0];
    const float* w    = (const float*)d_in[1];
    const float* csum = (const float*)d_in[2];
    const float* cnum = (const float*)d_in[3];
    float* out = (float*)d_out;

    dim3 grid(BATCH / BT, NCLS);   // batch tiles fastest -> same-class WGs co-resident (L2 reuse of weight slice)
    duq_rbf_kernel<<<grid, 256, 0, stream>>>(x, w, csum, cnum, out);
}